// GNNActor_1752346657367
// MI455X (gfx1250) — compile-verified
//
#include <hip/hip_runtime.h>
#include <math.h>

// Problem constants (match reference)
#define BB   2048
#define ACTN 14
#define CC   128
#define HH   32
#define NN   (BB*ACTN)        // 28672 nodes
#define EE   (BB*ACTN*ACTN)   // 401408 edges
#define PP   (ACTN*ACTN)      // 196 pairs
#define ROWS (BB*PP)          // 401408 MLP rows
#define PAD  33               // LDS row pad to dodge bank conflicts

typedef __attribute__((ext_vector_type(2))) float v2f;
typedef __attribute__((ext_vector_type(8))) float v8f;

static __device__ __forceinline__ v8f wmma_f32_k4(v2f a, v2f b, v8f c) {
  // D = A(16x4,f32) * B(4x16,f32) + C(16x16,f32)
  return __builtin_amdgcn_wmma_f32_16x16x4_f32(false, a, false, b, (short)0, c,
                                               false, false);
}

// ---------------------------------------------------------------- zero fill
__global__ void k_zero(float* __restrict__ p, size_t n) {
  for (size_t i = (size_t)blockIdx.x * blockDim.x + threadIdx.x; i < n;
       i += (size_t)gridDim.x * blockDim.x)
    p[i] = 0.0f;
}

// ---------------------------------------------------------------- degrees
__global__ void k_degree(const int* __restrict__ dst, float* __restrict__ deg) {
  int e = blockIdx.x * 256 + threadIdx.x;
  if (e < EE) unsafeAtomicAdd(&deg[dst[e]], 1.0f);
}

// ------------------------------------------------- xw = state @ conv_W (WMMA)
// block = 256 threads = 8 waves; blockIdx.x = 16-row tile; wave = 16-col tile
__global__ void k_xw_gemm(const float* __restrict__ X,
                          const float* __restrict__ W,
                          float* __restrict__ XW) {
  const int wave = threadIdx.x >> 5;
  const int lane = threadIdx.x & 31;
  const int lrow = lane & 15;
  const int hk   = (lane >> 4) * 2;       // K offset 0 or 2 per lane half
  const int rm   = blockIdx.x * 16;       // output row base
  const int cn   = wave * 16;             // output col base (8 waves * 16 = 128)

  v8f acc = {};
  for (int kb = 0; kb < CC; kb += 4) {
    const int k0 = kb + hk;
    v2f a, b;
    const float* ap = X + (size_t)(rm + lrow) * CC + k0;
    a.x = ap[0]; a.y = ap[1];
    const float* bp = W + (size_t)k0 * CC + cn + lrow;
    b.x = bp[0]; b.y = bp[CC];
    acc = wmma_f32_k4(a, b, acc);
  }
  // D layout: VGPR r -> (M=r, N=lane) lanes 0-15 ; (M=r+8, N=lane-16) lanes 16-31
  const int n     = cn + lrow;
  const int mbase = rm + ((lane >> 4) << 3);
#pragma unroll
  for (int r = 0; r < 8; ++r)
    XW[(size_t)(mbase + r) * CC + n] = acc[r];
}

// -------------------------------------------- agg[dst] += xw[src]*norm (edges)
// one wave per edge; each lane covers 4 channels
__global__ void k_edge_scatter(const int* __restrict__ src,
                               const int* __restrict__ dst,
                               const float* __restrict__ xw,
                               const float* __restrict__ deg,
                               float* __restrict__ agg) {
  int e = blockIdx.x * (blockDim.x >> 5) + (threadIdx.x >> 5);
  if (e >= EE) return;
  int lane = threadIdx.x & 31;
  int s = src[e], d = dst[e];
  float norm = rsqrtf(deg[s] + 1.0f) * rsqrtf(deg[d] + 1.0f);
  const float4* xs = (const float4*)(xw + (size_t)s * CC);
  float4 v = xs[lane];
  float* ad = agg + (size_t)d * CC + lane * 4;
  unsafeAtomicAdd(ad + 0, v.x * norm);
  unsafeAtomicAdd(ad + 1, v.y * norm);
  unsafeAtomicAdd(ad + 2, v.z * norm);
  unsafeAtomicAdd(ad + 3, v.w * norm);
}

// ------------------------- xfeat = relu(agg + xw*dinv2 + b) + state (in agg)
__global__ void k_finalize(const float* __restrict__ state,
                           const float* __restrict__ xw,
                           const float* __restrict__ convb,
                           const float* __restrict__ deg,
                           float* __restrict__ agg) {
  size_t i = (size_t)blockIdx.x * 256 + threadIdx.x;   // over N*C exactly
  int node = (int)(i >> 7);
  int c    = (int)(i & (CC - 1));
  float dinv2 = 1.0f / (deg[node] + 1.0f);
  float v = agg[i] + xw[i] * dinv2 + convb[c];
  v = fmaxf(v, 0.0f);
  agg[i] = v + state[i];
}

// ------------------------------------------------------- edge-pair MLP + heads
// block = 256 threads = 8 waves; each wave owns 16 rows of the 401408
__global__ void k_mlp(const float* __restrict__ xfeat,
                      const float* __restrict__ l1W, const float* __restrict__ l1b,
                      const float* __restrict__ l2W, const float* __restrict__ l2b,
                      const float* __restrict__ muW, const float* __restrict__ mub,
                      const float* __restrict__ sgW, const float* __restrict__ sgb,
                      float* __restrict__ out) {
  __shared__ float sW1[2 * CC * HH];   // 256x32
  __shared__ float sW2[HH * HH];       // 32x32
  __shared__ float sH[8][16 * PAD];    // per-wave 16x32 scratch (padded)

  for (int i = threadIdx.x; i < 2 * CC * HH; i += 256) sW1[i] = l1W[i];
  for (int i = threadIdx.x; i < HH * HH; i += 256)     sW2[i] = l2W[i];
  __syncthreads();

  const int wave = threadIdx.x >> 5;
  const int lane = threadIdx.x & 31;
  const int lrow = lane & 15;
  const int hk   = (lane >> 4) * 2;
  const int mrow = (lane >> 4) << 3;          // D-layout row offset 0/8

  const size_t rowbase = (size_t)blockIdx.x * 128 + (size_t)wave * 16;
  const size_t myrow   = rowbase + lrow;      // ROWS divisible by 128 -> in range
  const int b_ = (int)(myrow / PP);
  const int p  = (int)(myrow % PP);
  const float* srcRow = xfeat + (size_t)(b_ * ACTN + p / ACTN) * CC;
  const float* dstRow = xfeat + (size_t)(b_ * ACTN + p % ACTN) * CC;

  // ---- GEMM1: H1 = leaky(ef @ W1 + b1), ef = [src||dst] (K = 256)
  v8f acc0 = {}, acc1 = {};
  for (int kb = 0; kb < 2 * CC; kb += 4) {
    const int k0 = kb + hk;                   // never straddles the 128 boundary
    const float* r0 = (k0 < CC) ? (srcRow + k0) : (dstRow + (k0 - CC));
    v2f a;  a.x = r0[0];  a.y = r0[1];
    const float* w = &sW1[k0 * HH + lrow];
    v2f b0; b0.x = w[0];  b0.y = w[HH];
    v2f b1; b1.x = w[16]; b1.y = w[HH + 16];
    acc0 = wmma_f32_k4(a, b0, acc0);
    acc1 = wmma_f32_k4(a, b1, acc1);
  }
  {
    float bias0 = l1b[lrow], bias1 = l1b[16 + lrow];
    float* Hw = &sH[wave][0];
#pragma unroll
    for (int r = 0; r < 8; ++r) {
      float v0 = acc0[r] + bias0; v0 = v0 > 0.0f ? v0 : 0.01f * v0;
      float v1 = acc1[r] + bias1; v1 = v1 > 0.0f ? v1 : 0.01f * v1;
      Hw[(mrow + r) * PAD + lrow]      = v0;
      Hw[(mrow + r) * PAD + 16 + lrow] = v1;
    }
  }
  __syncthreads();

  // ---- GEMM2: H2 = leaky(H1 @ W2 + b2) (K = 32)
  float* Hw = &sH[wave][0];
  v8f c0 = {}, c1 = {};
  for (int kb = 0; kb < HH; kb += 4) {
    const int k0 = kb + hk;
    v2f a;  a.x = Hw[lrow * PAD + k0]; a.y = Hw[lrow * PAD + k0 + 1];
    const float* w = &sW2[k0 * HH + lrow];
    v2f b0; b0.x = w[0];  b0.y = w[HH];
    v2f b1; b1.x = w[16]; b1.y = w[HH + 16];
    c0 = wmma_f32_k4(a, b0, c0);
    c1 = wmma_f32_k4(a, b1, c1);
  }
  __syncthreads();
  {
    float bias0 = l2b[lrow], bias1 = l2b[16 + lrow];
#pragma unroll
    for (int r = 0; r < 8; ++r) {
      float v0 = c0[r] + bias0; v0 = v0 > 0.0f ? v0 : 0.01f * v0;
      float v1 = c1[r] + bias1; v1 = v1 > 0.0f ? v1 : 0.01f * v1;
      Hw[(mrow + r) * PAD + lrow]      = v0;
      Hw[(mrow + r) * PAD + 16 + lrow] = v1;
    }
  }
  __syncthreads();

  // ---- heads: lanes 0-15 -> mu row lrow ; lanes 16-31 -> std row lrow
  const float* hrow = &Hw[lrow * PAD];
  const float* wv   = (lane < 16) ? muW : sgW;
  float dot = 0.0f;
#pragma unroll
  for (int k = 0; k < HH; ++k) dot += hrow[k] * wv[k];

  size_t orow = rowbase + lrow;
  if (lane < 16) {
    float x = dot + mub[0] + 1e-10f;
    float sp = (x > 20.0f) ? x : log1pf(expf(x));   // softplus
    out[orow] = sp;                                  // out[0][b][p]
  } else {
    float ls = dot + sgb[0];
    ls = fminf(fmaxf(ls, -20.0f), 2.0f);
    out[(size_t)BB * PP + orow] = expf(ls);          // out[1][b][p]
  }
}

// =============================================================== launch glue
extern "C" void kernel_launch(void* const* d_in, const int* in_sizes, int n_in,
                              void* d_out, int out_size, void* d_ws, size_t ws_size,
                              hipStream_t stream) {
  const float* state = (const float*)d_in[0];
  const float* convW = (const float*)d_in[1];
  const float* convb = (const float*)d_in[2];
  const float* l1W   = (const float*)d_in[3];
  const float* l1b   = (const float*)d_in[4];
  const float* l2W   = (const float*)d_in[5];
  const float* l2b   = (const float*)d_in[6];
  const float* muW   = (const float*)d_in[7];
  const float* mub   = (const float*)d_in[8];
  const float* sgW   = (const float*)d_in[9];
  const float* sgb   = (const float*)d_in[10];
  const int*   eidx  = (const int*)d_in[11];
  const int*   esrc  = eidx;
  const int*   edst  = eidx + EE;
  float*       out   = (float*)d_out;

  // workspace: deg[N] | xw[N*C] | agg[N*C]   (~29.5 MB)
  float* deg = (float*)d_ws;
  float* xw  = deg + NN;
  float* agg = xw + (size_t)NN * CC;

  k_zero<<<512, 256, 0, stream>>>(deg, (size_t)NN);
  k_zero<<<2048, 256, 0, stream>>>(agg, (size_t)NN * CC);
  k_degree<<<(EE + 255) / 256, 256, 0, stream>>>(edst, deg);
  k_xw_gemm<<<NN / 16, 256, 0, stream>>>(state, convW, xw);
  k_edge_scatter<<<(EE + 7) / 8, 256, 0, stream>>>(esrc, edst, xw, deg, agg);
  k_finalize<<<((size_t)NN * CC) / 256, 256, 0, stream>>>(state, xw, convb, deg, agg);
  k_mlp<<<ROWS / 128, 256, 0, stream>>>(agg, l1W, l1b, l2W, l2b,
                                        muW, mub, sgW, sgb, out);
}